// MixResNetGNN_14697378087222
// MI455X (gfx1250) — compile-verified
//
#include <hip/hip_runtime.h>

typedef __attribute__((ext_vector_type(2))) float v2f;
typedef __attribute__((ext_vector_type(8))) float v8f;

#define GN 50000
#define GE 800000
#define GIN 256
#define GHID 128
#define GOUT 64
#define GLAYERS 3
#define LN_EPS 1e-5f

__device__ __forceinline__ void atomic_add_f32(float* p, float v) {
  __hip_atomic_fetch_add(p, v, __ATOMIC_RELAXED, __HIP_MEMORY_SCOPE_AGENT);
}

// ---------------- graph normalization ----------------

__global__ void k_fill(float* __restrict__ p, float v, int n) {
  int i = blockIdx.x * blockDim.x + threadIdx.x;
  if (i < n) p[i] = v;
}

__global__ void k_deg_edges(const int* __restrict__ dst, float* __restrict__ deg, int e) {
  int i = blockIdx.x * blockDim.x + threadIdx.x;
  if (i < e) atomic_add_f32(deg + dst[i], 1.0f);
}

__global__ void k_dinv(float* __restrict__ deg, int n) {
  int i = blockIdx.x * blockDim.x + threadIdx.x;
  if (i < n) {
    float d = deg[i];
    deg[i] = (d > 0.0f) ? rsqrtf(d) : 0.0f;
  }
}

__global__ void k_norm(const int* __restrict__ src, const int* __restrict__ dst,
                       const float* __restrict__ dinv, float* __restrict__ nrm, int e) {
  int i = blockIdx.x * blockDim.x + threadIdx.x;
  if (i < e) nrm[i] = dinv[src[i]] * dinv[dst[i]];
}

// ---------------- fp32 WMMA GEMM: C[M,NC] = A[M,K] @ W[NC,K]^T (+bias) ----------------
// One wave computes a 64x16 tile (4 M-tiles sharing one B fragment per K-step of 4).
// Software-pipelined: next k-step's 5 fragment loads issue before this step's WMMAs,
// so the scheduler can emit partial s_wait_loadcnt and overlap matrix ops with loads.
// Optional fused epilogue: Y[row,col] = ybias[col] + C[row,col] * dinv[row]^2
// (GCN self-loop + conv-bias init), saving a full re-read of C.

__global__ __launch_bounds__(256) void k_gemm_wmma(
    const float* __restrict__ A, const float* __restrict__ W,
    const float* __restrict__ bias, float* __restrict__ C,
    const float* __restrict__ dinv, float* __restrict__ Y,
    const float* __restrict__ ybias,
    int M, int K, int NC) {
  const int lane = threadIdx.x & 31;
  const int wid = blockIdx.x * (blockDim.x >> 5) + (threadIdx.x >> 5);
  const int tiles_n = NC >> 4;
  const int tm = wid / tiles_n;
  const int tn = wid - tm * tiles_n;
  const int row_base = tm * 64;
  if (row_base >= M) return;                 // wave-uniform exit: EXEC stays full
  const int n0 = tn * 16;
  const int lgrp = lane >> 4;                // 0 | 1
  const int lmod = lane & 15;
  const int col = n0 + lmod;
  const int kk = lgrp << 1;                  // this lane's K-offset within a step

  // Clamp tail rows so loads stay non-divergent (stores are masked below).
  const float* Ar[4];
#pragma unroll
  for (int t = 0; t < 4; ++t) {
    int r = row_base + t * 16 + lmod;
    r = (r < M) ? r : (M - 1);
    Ar[t] = A + (size_t)r * K;
  }
  const float* Wr = W + (size_t)col * K;

  v8f acc[4] = {v8f{}, v8f{}, v8f{}, v8f{}};

  // prologue: fragments for k0 = 0
  float2 bw = *(const float2*)(Wr + kk);
  float2 aw[4];
#pragma unroll
  for (int t = 0; t < 4; ++t) aw[t] = *(const float2*)(Ar[t] + kk);

  for (int k0 = 0; k0 < K; k0 += 4) {
    // prefetch next k-step (wrap to k=0 on last trip: harmless, stays in bounds)
    const int knext = (k0 + 4 < K) ? (k0 + 4 + kk) : kk;
    float2 bw_n = *(const float2*)(Wr + knext);
    float2 aw_n[4];
#pragma unroll
    for (int t = 0; t < 4; ++t) aw_n[t] = *(const float2*)(Ar[t] + knext);

    v2f bv; bv.x = bw.x; bv.y = bw.y;
#pragma unroll
    for (int t = 0; t < 4; ++t) {
      v2f av; av.x = aw[t].x; av.y = aw[t].y;
      acc[t] = __builtin_amdgcn_wmma_f32_16x16x4_f32(
          /*neg_a=*/false, av, /*neg_b=*/false, bv,
          /*c_mod=*/(short)0, acc[t], /*reuse_a=*/false, /*reuse_b=*/false);
    }

    bw = bw_n;
#pragma unroll
    for (int t = 0; t < 4; ++t) aw[t] = aw_n[t];
  }

  const float bz = bias ? bias[col] : 0.0f;
  const float yb = ybias ? ybias[col] : 0.0f;
#pragma unroll
  for (int t = 0; t < 4; ++t) {
#pragma unroll
    for (int v = 0; v < 8; ++v) {
      int row = row_base + t * 16 + (lgrp << 3) + v;
      if (row < M) {
        float val = acc[t][v] + bz;
        C[(size_t)row * NC + col] = val;
        if (Y) {
          float dv = dinv[row];
          Y[(size_t)row * NC + col] = yb + val * dv * dv;
        }
      }
    }
  }
}

// ---------------- edge scatter: one wave per edge, float4 per lane ----------------

__global__ __launch_bounds__(256) void k_scatter(
    const int* __restrict__ src, const int* __restrict__ dst,
    const float* __restrict__ nrm, const float* __restrict__ ht,
    float* __restrict__ y) {
  int tid = blockIdx.x * blockDim.x + threadIdx.x;
  int e = tid >> 5;
  if (e >= GE) return;
  int c = (tid & 31) << 2;
  int s = src[e], d = dst[e];
  float nm = nrm[e];
  float4 hv = *(const float4*)(ht + (size_t)s * GHID + c);
  float* yp = y + (size_t)d * GHID + c;
  atomic_add_f32(yp + 0, hv.x * nm);
  atomic_add_f32(yp + 1, hv.y * nm);
  atomic_add_f32(yp + 2, hv.z * nm);
  atomic_add_f32(yp + 3, hv.w * nm);
}

// ---------------- LayerNorm + ReLU/residual mix: one wave32 per node ----------------

__global__ __launch_bounds__(256) void k_ln_mix(
    const float* __restrict__ y, const float* __restrict__ g,
    const float* __restrict__ b, float* __restrict__ h) {
  int lane = threadIdx.x & 31;
  int node = blockIdx.x * (blockDim.x >> 5) + (threadIdx.x >> 5);
  if (node >= GN) return;
  int c = lane << 2;
  float4 yv = *(const float4*)(y + (size_t)node * GHID + c);

  float s = yv.x + yv.y + yv.z + yv.w;
#pragma unroll
  for (int off = 16; off > 0; off >>= 1) s += __shfl_xor(s, off, 32);
  float mu = s * (1.0f / GHID);

  float dx = yv.x - mu, dy = yv.y - mu, dz = yv.z - mu, dw = yv.w - mu;
  float vs = dx * dx + dy * dy + dz * dz + dw * dw;
#pragma unroll
  for (int off = 16; off > 0; off >>= 1) vs += __shfl_xor(vs, off, 32);
  float rs = rsqrtf(vs * (1.0f / GHID) + LN_EPS);

  float4 gv = *(const float4*)(g + c);
  float4 bv = *(const float4*)(b + c);
  float4 hv = *(const float4*)(h + (size_t)node * GHID + c);

  float z0 = dx * rs * gv.x + bv.x;
  float z1 = dy * rs * gv.y + bv.y;
  float z2 = dz * rs * gv.z + bv.z;
  float z3 = dw * rs * gv.w + bv.w;
  float4 o;
  o.x = 0.5f * fmaxf(z0, 0.0f) + 0.5f * yv.x + 0.5f * hv.x;
  o.y = 0.5f * fmaxf(z1, 0.0f) + 0.5f * yv.y + 0.5f * hv.y;
  o.z = 0.5f * fmaxf(z2, 0.0f) + 0.5f * yv.z + 0.5f * hv.z;
  o.w = 0.5f * fmaxf(z3, 0.0f) + 0.5f * yv.w + 0.5f * hv.w;
  *(float4*)(h + (size_t)node * GHID + c) = o;
}

// ---------------- launcher ----------------

extern "C" void kernel_launch(void* const* d_in, const int* in_sizes, int n_in,
                              void* d_out, int out_size, void* d_ws, size_t ws_size,
                              hipStream_t stream) {
  (void)in_sizes; (void)n_in; (void)out_size; (void)ws_size;
  const float* x      = (const float*)d_in[0];
  const int*   ei     = (const int*)d_in[1];
  const int*   src    = ei;
  const int*   dst    = ei + GE;
  const float* W_in   = (const float*)d_in[2];
  const float* b_in   = (const float*)d_in[3];
  const float* W_conv = (const float*)d_in[4];
  const float* b_conv = (const float*)d_in[5];
  const float* ln_g   = (const float*)d_in[6];
  const float* ln_b   = (const float*)d_in[7];
  const float* W_out  = (const float*)d_in[8];
  const float* b_out  = (const float*)d_in[9];
  float* out = (float*)d_out;

  char* ws = (char*)d_ws;
  size_t off = 0;
  auto alloc = [&](size_t bytes) -> void* {
    void* p = ws + off;
    off += (bytes + 255) & ~(size_t)255;
    return p;
  };
  float* h    = (float*)alloc((size_t)GN * GHID * sizeof(float));
  float* ht   = (float*)alloc((size_t)GN * GHID * sizeof(float));
  float* y    = (float*)alloc((size_t)GN * GHID * sizeof(float));
  float* dinv = (float*)alloc((size_t)GN * sizeof(float));
  float* nrm  = (float*)alloc((size_t)GE * sizeof(float));

  // degrees (self-loop contributes 1), dinv, per-edge norm
  k_fill<<<(GN + 255) / 256, 256, 0, stream>>>(dinv, 1.0f, GN);
  k_deg_edges<<<(GE + 255) / 256, 256, 0, stream>>>(dst, dinv, GE);
  k_dinv<<<(GN + 255) / 256, 256, 0, stream>>>(dinv, GN);
  k_norm<<<(GE + 255) / 256, 256, 0, stream>>>(src, dst, dinv, nrm, GE);

  const int m64 = (GN + 63) / 64;

  // in_lin: h = x @ W_in^T + b_in
  {
    int waves = m64 * (GHID / 16);
    k_gemm_wmma<<<(waves + 7) / 8, 256, 0, stream>>>(
        x, W_in, b_in, h, nullptr, nullptr, nullptr, GN, GIN, GHID);
  }

  for (int l = 0; l < GLAYERS; ++l) {
    // ht = h @ W_conv[l]^T ; fused: y = b_conv[l] + ht * dinv^2 (self-loop + bias)
    {
      int waves = m64 * (GHID / 16);
      k_gemm_wmma<<<(waves + 7) / 8, 256, 0, stream>>>(
          h, W_conv + (size_t)l * GHID * GHID, nullptr, ht,
          dinv, y, b_conv + (size_t)l * GHID, GN, GHID, GHID);
    }
    // scatter-add edge messages into y
    k_scatter<<<((GE * 32) + 255) / 256, 256, 0, stream>>>(src, dst, nrm, ht, y);
    // h = 0.5*relu(LN(y)) + 0.5*y + 0.5*h   (in place on h)
    k_ln_mix<<<(GN + 7) / 8, 256, 0, stream>>>(
        y, ln_g + (size_t)l * GHID, ln_b + (size_t)l * GHID, h);
  }

  // out = h @ W_out^T + b_out
  {
    int waves = m64 * (GOUT / 16);
    k_gemm_wmma<<<(waves + 7) / 8, 256, 0, stream>>>(
        h, W_out, b_out, out, nullptr, nullptr, nullptr, GN, GHID, GOUT);
  }
}